// GAT_21071109554678
// MI455X (gfx1250) — compile-verified
//
#include <hip/hip_runtime.h>
#include <hip/hip_bf16.h>

#define N_NODES  50000
#define N_EDGES  800000
#define IN_FEAT  128
#define HEADS    4
#define UNITS    16
#define OUTF     (HEADS * UNITS)   /* 64 */
#define LEAKY_A  0.2f

typedef __attribute__((ext_vector_type(2))) float        v2f;
typedef __attribute__((ext_vector_type(8))) float        v8f;
typedef __attribute__((ext_vector_type(4))) unsigned int u32x4;
typedef __attribute__((ext_vector_type(8))) int          i32x8;
typedef __attribute__((ext_vector_type(4))) int          i32x4;

#if defined(__HIP_DEVICE_COMPILE__) && __has_builtin(__builtin_amdgcn_tensor_load_to_lds)
#define HAVE_TDM 1
#endif
#if defined(__HIP_DEVICE_COMPILE__) && __has_builtin(__builtin_amdgcn_wmma_f32_16x16x4_f32)
#define HAVE_WMMA_F32 1
#endif

// ---------------------------------------------------------------------------
// Helpers
// ---------------------------------------------------------------------------

// Low 32 bits of a flat LDS pointer == LDS byte offset.
__device__ __forceinline__ unsigned lds_off(const void* p) {
  return (unsigned)(size_t)p;
}

#if defined(HAVE_TDM)
// Build a 2D Tensor-DMA descriptor (D#) and issue TENSOR_LOAD_TO_LDS.
// dims/strides/tiles are in 4-byte elements (data_size=2 -> 4B).
// clang-23 lane: 6-arg builtin (u32x4, i32x8, i32x4, i32x4, i32x8, i32 cpol).
__device__ __forceinline__ void tdm_load_2d(unsigned lds_addr, const void* gaddr,
                                            unsigned tdim0, unsigned tdim1,
                                            unsigned stride0,
                                            unsigned tile0, unsigned tile1) {
  unsigned long long ga = (unsigned long long)gaddr;
  u32x4 g0;
  g0.x = 1u;                                            // count=1, is_restore=0
  g0.y = lds_addr;                                      // lds_addr [63:32]
  g0.z = (unsigned)(ga & 0xFFFFFFFFull);                // global_addr [95:64]
  g0.w = (unsigned)((ga >> 32) & 0x01FFFFFFull)         // global_addr [120:96]
         | (2u << 30);                                  // type=2 ("image")
  i32x8 g1;
  g1[0] = (int)(2u << 16);                              // data_size=4B, no multicast
  g1[1] = (int)((tdim0 & 0xFFFFu) << 16);               // tensor_dim0[15:0]
  g1[2] = (int)((tdim0 >> 16) | ((tdim1 & 0xFFFFu) << 16));
  g1[3] = (int)((tdim1 >> 16) | ((tile0 & 0xFFFFu) << 16)); // tile_dim0
  g1[4] = (int)(tile1 & 0xFFFFu);                       // tile_dim1 (tile_dim2=0)
  g1[5] = (int)stride0;                                 // tensor_dim0_stride[31:0]
  g1[6] = 0;
  g1[7] = 0;
  i32x4 gz4 = {0, 0, 0, 0};                             // 2D: groups 2/3 unused
  i32x8 gz8 = {0, 0, 0, 0, 0, 0, 0, 0};
  __builtin_amdgcn_tensor_load_to_lds(g0, g1, gz4, gz4, gz8, 0);
}
#endif

// Order-preserving float<->u32 encoding so segment-max can use integer atomicMax.
__device__ __forceinline__ unsigned fenc(float f) {
  unsigned b = __float_as_uint(f);
  return (b & 0x80000000u) ? ~b : (b | 0x80000000u);
}
__device__ __forceinline__ float fdec(unsigned u) {
  return __uint_as_float((u & 0x80000000u) ? (u & 0x7FFFFFFFu) : ~u);
}
#define ENC_NEG_INF 0x007FFFFFu   /* fenc(-inf) */

// ---------------------------------------------------------------------------
// Kernel 0: re-initialize accumulation buffers (graph replays re-use them)
// ---------------------------------------------------------------------------
__global__ void k_init(float* __restrict__ out, unsigned* __restrict__ maxenc,
                       float* __restrict__ denom) {
  int t = blockIdx.x * blockDim.x + threadIdx.x;
  if (t < N_NODES * OUTF) out[t] = 0.0f;
  if (t < N_NODES * HEADS) { maxenc[t] = ENC_NEG_INF; denom[t] = 0.0f; }
}

// ---------------------------------------------------------------------------
// Kernel 1: h = x @ W via V_WMMA_F32_16X16X4_F32, A/B staged in LDS by the TDM
//   block = 128 threads (4 waves); block owns 16 rows, wave w owns N-tile w*16
// ---------------------------------------------------------------------------
__global__ __launch_bounds__(128) void k_gemm_wmma(const float* __restrict__ x,
                                                   const float* __restrict__ W,
                                                   float* __restrict__ hbuf) {
  __shared__ float As[16 * IN_FEAT];    // 8 KB: x tile, dense 16x128
  __shared__ float Ws[IN_FEAT * OUTF];  // 32 KB: full W, dense 128x64
  const int m0   = blockIdx.x * 16;
  const int lane = (int)(threadIdx.x & 31u);
  const int wave = (int)(threadIdx.x >> 5u);

#if defined(HAVE_TDM)
  if (wave == 0) {                      // one DMA pair per block (TDM ignores EXEC)
    tdm_load_2d(lds_off(As), x + (size_t)m0 * IN_FEAT,
                /*tdim0=*/IN_FEAT, /*tdim1=*/16, /*stride0=*/IN_FEAT,
                /*tile0=*/IN_FEAT, /*tile1=*/16);
    tdm_load_2d(lds_off(Ws), W,
                /*tdim0=*/OUTF, /*tdim1=*/IN_FEAT, /*stride0=*/OUTF,
                /*tile0=*/OUTF, /*tile1=*/IN_FEAT);
    __builtin_amdgcn_s_wait_tensorcnt(0);
  }
#else
  for (int i = (int)threadIdx.x; i < 16 * IN_FEAT; i += 128)
    As[i] = x[(size_t)m0 * IN_FEAT + i];
  for (int i = (int)threadIdx.x; i < IN_FEAT * OUTF; i += 128)
    Ws[i] = W[i];
#endif
  __syncthreads();

  const int n0    = wave * 16;
  const int rc    = lane & 15;          // A row / B col within tile
  const int khalf = (lane >> 4) * 2;    // lanes 0-15: K={0,1}; lanes 16-31: K={2,3}
  v8f acc = {};

#pragma unroll
  for (int k0 = 0; k0 < IN_FEAT; k0 += 4) {
    v2f a, b;
    a.x = As[rc * IN_FEAT + k0 + khalf];
    a.y = As[rc * IN_FEAT + k0 + khalf + 1];
    b.x = Ws[(k0 + khalf) * OUTF + n0 + rc];
    b.y = Ws[(k0 + khalf + 1) * OUTF + n0 + rc];
#if defined(HAVE_WMMA_F32)
    acc = __builtin_amdgcn_wmma_f32_16x16x4_f32(
        /*neg_a=*/false, a, /*neg_b=*/false, b,
        /*c_mod=*/(short)0, acc, /*reuse_a=*/false, /*reuse_b=*/false);
#else
    acc[0] += a.x * b.x;                // host-pass placeholder, never executed
#endif
  }

  // C/D layout: VGPR i -> M=i (lanes 0-15) or M=8+i (lanes 16-31); N = n0 + rc
  const int mbase = m0 + (lane >> 4) * 8;
#pragma unroll
  for (int i = 0; i < 8; ++i)
    hbuf[(size_t)(mbase + i) * OUTF + n0 + rc] = acc[i];
}

// ---------------------------------------------------------------------------
// Kernel 2: per-node attention scalars  s1[n,h]=a1_h·h[n,h,:], s2 likewise
// ---------------------------------------------------------------------------
__global__ void k_node_scores(const float* __restrict__ hbuf,
                              const float* __restrict__ aw1,
                              const float* __restrict__ aw2,
                              float* __restrict__ s1, float* __restrict__ s2) {
  int t = blockIdx.x * blockDim.x + threadIdx.x;
  if (t >= N_NODES * HEADS) return;
  int n = t >> 2, h = t & 3;
  const float* hp = hbuf + (size_t)n * OUTF + h * UNITS;
  float a1 = 0.0f, a2 = 0.0f;
#pragma unroll
  for (int u = 0; u < UNITS; ++u) {
    float v = hp[u];
    a1 += aw1[h * UNITS + u] * v;
    a2 += aw2[h * UNITS + u] * v;
  }
  s1[t] = a1;
  s2[t] = a2;
}

// ---------------------------------------------------------------------------
// Kernel 3: edge logits + LeakyReLU + segment-max (u32-encoded atomicMax)
// ---------------------------------------------------------------------------
__global__ void k_edge_att(const int* __restrict__ ei,
                           const float* __restrict__ s1,
                           const float* __restrict__ s2,
                           const float* __restrict__ hbuf,
                           float* __restrict__ attbuf,
                           unsigned* __restrict__ maxenc) {
  int t = blockIdx.x * blockDim.x + threadIdx.x;
  if (t >= N_EDGES * HEADS) return;
  int e = t >> 2, h = t & 3;
  int snd = ei[2 * e], rcv = ei[2 * e + 1];
  float a = s1[snd * HEADS + h] + s2[rcv * HEADS + h];
  a = (a > 0.0f) ? a : LEAKY_A * a;
  attbuf[t] = a;
  atomicMax(&maxenc[rcv * HEADS + h], fenc(a));
  if (h == 0)  // warm L2 with the sender row the aggregation pass will gather
    __builtin_prefetch(hbuf + (size_t)snd * OUTF, 0, 1);
}

// ---------------------------------------------------------------------------
// Kernel 4: e = exp(att - max[rcv]); denom[rcv] += e   (in-place over attbuf)
// ---------------------------------------------------------------------------
__global__ void k_edge_exp(const int* __restrict__ ei,
                           float* __restrict__ attbuf,
                           const unsigned* __restrict__ maxenc,
                           float* __restrict__ denom) {
  int t = blockIdx.x * blockDim.x + threadIdx.x;
  if (t >= N_EDGES * HEADS) return;
  int e = t >> 2, h = t & 3;
  int rcv = ei[2 * e + 1];
  float m  = fdec(maxenc[rcv * HEADS + h]);
  float ex = __expf(attbuf[t] - m);
  attbuf[t] = ex;
  atomicAdd(&denom[rcv * HEADS + h], ex);
}

// ---------------------------------------------------------------------------
// Kernel 5: out[rcv,:] += (e/denom[rcv]) * h[snd,:]   (64 lanes per edge)
// ---------------------------------------------------------------------------
__global__ void k_edge_agg(const int* __restrict__ ei,
                           const float* __restrict__ ebuf,
                           const float* __restrict__ denom,
                           const float* __restrict__ hbuf,
                           float* __restrict__ out) {
  int t = blockIdx.x * blockDim.x + threadIdx.x;
  if (t >= N_EDGES * OUTF) return;
  int e = t >> 6, c = t & 63, h = c >> 4;
  int snd = ei[2 * e], rcv = ei[2 * e + 1];
  float alpha = ebuf[e * HEADS + h] / denom[rcv * HEADS + h];
  atomicAdd(&out[(size_t)rcv * OUTF + c], alpha * hbuf[(size_t)snd * OUTF + c]);
}

// ---------------------------------------------------------------------------
// Host launcher
// ---------------------------------------------------------------------------
extern "C" void kernel_launch(void* const* d_in, const int* in_sizes, int n_in,
                              void* d_out, int out_size, void* d_ws, size_t ws_size,
                              hipStream_t stream) {
  (void)in_sizes; (void)n_in; (void)out_size; (void)ws_size;
  const float* x   = (const float*)d_in[0];
  const int*   ei  = (const int*)d_in[1];
  const float* W   = (const float*)d_in[2];
  const float* aw1 = (const float*)d_in[3];
  const float* aw2 = (const float*)d_in[4];
  float* out = (float*)d_out;

  // Workspace layout (~28.8 MB)
  float*    hbuf   = (float*)d_ws;                              // 50000*64
  float*    s1     = hbuf + (size_t)N_NODES * OUTF;             // 50000*4
  float*    s2     = s1 + (size_t)N_NODES * HEADS;              // 50000*4
  float*    denom  = s2 + (size_t)N_NODES * HEADS;              // 50000*4
  unsigned* maxenc = (unsigned*)(denom + (size_t)N_NODES * HEADS); // 50000*4
  float*    attbuf = (float*)(maxenc + (size_t)N_NODES * HEADS);   // 800000*4

  k_init<<<dim3((N_NODES * OUTF + 255) / 256), dim3(256), 0, stream>>>(out, maxenc, denom);
  k_gemm_wmma<<<dim3(N_NODES / 16), dim3(128), 0, stream>>>(x, W, hbuf);
  k_node_scores<<<dim3((N_NODES * HEADS + 255) / 256), dim3(256), 0, stream>>>(hbuf, aw1, aw2, s1, s2);
  k_edge_att<<<dim3((N_EDGES * HEADS + 255) / 256), dim3(256), 0, stream>>>(ei, s1, s2, hbuf, attbuf, maxenc);
  k_edge_exp<<<dim3((N_EDGES * HEADS + 255) / 256), dim3(256), 0, stream>>>(ei, attbuf, maxenc, denom);
  k_edge_agg<<<dim3((N_EDGES * OUTF + 255) / 256), dim3(256), 0, stream>>>(ei, attbuf, denom, hbuf, out);
}